// NonMaximumSuppression_87943750352914
// MI455X (gfx1250) — compile-verified
//
#include <hip/hip_runtime.h>
#include <stdint.h>

// ---------------------------------------------------------------------------
// NMS for MI455X (gfx1250): rank-select top-1000 (stable, exact vs jnp.argsort)
// then greedy NMS with a full 1000x1000 suppression bit-matrix in LDS (148KB,
// exploiting CDNA5's 320KB/WGP) and a barrier-free single-wave32 serial scan.
// Async global->LDS (ASYNCcnt) staging where the toolchain exposes it.
// ---------------------------------------------------------------------------

#define NMS_B    16
#define NMS_P    20000
#define NMS_TOP  1000
#define NMS_THR  0.5f

#define K1_THREADS 256
#define K1_ILP     8
#define K1_SPAN    (K1_THREADS * K1_ILP)   // 2048 candidates per block
#define K1_TILE    1024                    // scores staged per LDS tile

#define K2_THREADS 1024
#define K2_WORDS   32                      // ceil(1000/32)

#if defined(__HIP_DEVICE_COMPILE__) && defined(__has_builtin)
#if __has_builtin(__builtin_amdgcn_global_load_async_to_lds_b128) && \
    __has_builtin(__builtin_amdgcn_s_wait_asynccnt)
#define NMS_ASYNC_LDS 1
#endif
#endif

// Monotone map: larger score -> SMALLER key (descending order), so that
// ascending rank over ((desc<<32)|index) reproduces stable argsort(-cls).
__device__ __forceinline__ unsigned mono_desc(float f) {
    unsigned u = __float_as_uint(f);
    unsigned asc = (u & 0x80000000u) ? ~u : (u | 0x80000000u);
    return ~asc;
}

#ifdef NMS_ASYNC_LDS
// Builtin signature (from hipcc diagnostic): first param is
//   '__vector_size__(4*sizeof(int)) int __device__ *'  (v4i in AS1, global src)
// second is the LDS destination (AS3).
typedef int v4i_t __attribute__((vector_size(16)));
typedef __attribute__((address_space(1))) v4i_t* gptr_v4i;
typedef __attribute__((address_space(3))) v4i_t* lptr_v4i;

__device__ __forceinline__ void async_copy_b128(const void* gsrc, void* ldst) {
    __builtin_amdgcn_global_load_async_to_lds_b128(
        (gptr_v4i)(uintptr_t)gsrc,   // global source (AS1)
        (lptr_v4i)ldst,              // LDS destination (AS3)
        0, 0);                       // imm offset, cpol
}
#endif

// ---------------------------------------------------------------------------
// Kernel 1: exact rank selection. grid = (ceil(P/2048), B), block = 256.
// Each thread owns 8 candidates; all 20000 keys stream through LDS once per
// block; one broadcast ds_load feeds 8 register-resident u64 comparisons.
// Rank < 1000 scatters score+box directly into sorted position in workspace.
// ---------------------------------------------------------------------------
__global__ __launch_bounds__(K1_THREADS)
void nms_topk_rank(const float* __restrict__ reg, const float* __restrict__ cls,
                   float* __restrict__ wsBox, float* __restrict__ wsCls)
{
    __shared__ float              sVal[K1_TILE];
    __shared__ unsigned long long sKey[K1_TILE];

    const int b   = blockIdx.y;
    const int tid = threadIdx.x;
    const float* __restrict__ clsB = cls + (size_t)b * NMS_P;

    int                idx[K1_ILP];
    float              sc [K1_ILP];
    unsigned long long key[K1_ILP];
    unsigned           cnt[K1_ILP];

    #pragma unroll
    for (int k = 0; k < K1_ILP; ++k) {
        const int i = blockIdx.x * K1_SPAN + k * K1_THREADS + tid;
        idx[k] = i;
        cnt[k] = 0u;
        if (i < NMS_P) {
            const float s = clsB[i];
            sc[k]  = s;
            key[k] = ((unsigned long long)mono_desc(s) << 32) | (unsigned)i;
            // warm L2 for the box gather at the end (global_prefetch_b8)
            __builtin_prefetch(reg + ((size_t)b * NMS_P + i) * 4, 0, 1);
        } else {
            sc[k]  = 0.f;
            key[k] = ~0ull;     // guarded out at the write below
        }
    }

    for (int base = 0; base < NMS_P; base += K1_TILE) {
        const int n = (NMS_P - base < K1_TILE) ? (NMS_P - base) : K1_TILE;

        // Stage a tile of scores into LDS (128b per lane). n is always a
        // multiple of 4 (20000 % 1024 = 544 = 136*4).
        const int e0 = tid * 4;
        if (e0 < n) {
#ifdef NMS_ASYNC_LDS
            async_copy_b128(clsB + base + e0, &sVal[e0]);
#else
            *reinterpret_cast<float4*>(&sVal[e0]) =
                *reinterpret_cast<const float4*>(clsB + base + e0);
#endif
        }
#ifdef NMS_ASYNC_LDS
        __builtin_amdgcn_s_wait_asynccnt(0);
#endif
        __syncthreads();

        // Convert staged scores to unique 64-bit sort keys.
        for (int e = tid; e < K1_TILE; e += K1_THREADS) {
            const int g = base + e;
            sKey[e] = (g < NMS_P)
                ? (((unsigned long long)mono_desc(sVal[e]) << 32) | (unsigned)g)
                : ~0ull;        // pad: never smaller than any real key
        }
        __syncthreads();

        // Rank accumulation: broadcast LDS read, 8 compares per key.
        #pragma unroll 4
        for (int e = 0; e < K1_TILE; ++e) {
            const unsigned long long kj = sKey[e];
            #pragma unroll
            for (int k = 0; k < K1_ILP; ++k)
                cnt[k] += (kj < key[k]) ? 1u : 0u;
        }
        __syncthreads();
    }

    // Ranks are a bijection -> each of the 1000 slots written exactly once.
    #pragma unroll
    for (int k = 0; k < K1_ILP; ++k) {
        if (idx[k] < NMS_P && cnt[k] < NMS_TOP) {
            const size_t dst = (size_t)b * NMS_TOP + cnt[k];
            wsCls[dst] = sc[k];
            const float4 bx = *reinterpret_cast<const float4*>(
                reg + ((size_t)b * NMS_P + idx[k]) * 4);
            *reinterpret_cast<float4*>(wsBox + dst * 4) = bx;
        }
    }
}

// ---------------------------------------------------------------------------
// Kernel 2: greedy NMS. grid = B, block = 1024.
// Phase 1: build the 1000x1000 IoU>=thr bit-matrix in LDS (125KB) in parallel.
// Phase 2: one wave32 does the 1000-step serial scan with zero barriers:
//          lane w owns removed word w; per step one shuffle + masked OR.
// ---------------------------------------------------------------------------
__global__ __launch_bounds__(K2_THREADS)
void nms_greedy(const float* __restrict__ wsBox, const float* __restrict__ wsCls,
                float* __restrict__ outReg, float* __restrict__ outCls)
{
    __shared__ float4   sBox[NMS_TOP];                 // 16000 B
    __shared__ float    sArea[NMS_TOP];                //  4000 B
    __shared__ unsigned sMat[NMS_TOP * K2_WORDS];      // 128000 B
    __shared__ unsigned sRem[K2_WORDS];                //   128 B

    const int b   = blockIdx.x;
    const int tid = threadIdx.x;

    if (tid < NMS_TOP) {
#ifdef NMS_ASYNC_LDS
        async_copy_b128(wsBox + ((size_t)b * NMS_TOP + tid) * 4, &sBox[tid]);
#else
        sBox[tid] = *reinterpret_cast<const float4*>(
            wsBox + ((size_t)b * NMS_TOP + tid) * 4);
#endif
    }
#ifdef NMS_ASYNC_LDS
    __builtin_amdgcn_s_wait_asynccnt(0);
#endif
    __syncthreads();

    if (tid < NMS_TOP) {
        const float4 bx = sBox[tid];
        sArea[tid] = (bx.z - bx.x) * (bx.w - bx.y);
    }
    __syncthreads();

    // Build suppression rows. Task mapping i = task % 1000 keeps the inner
    // sBox[j] reads uniform across a wave (LDS broadcast, conflict-free).
    for (int task = tid; task < NMS_TOP * K2_WORDS; task += K2_THREADS) {
        const int i = task % NMS_TOP;
        const int w = task / NMS_TOP;
        const float4 a  = sBox[i];
        const float  aa = sArea[i];
        unsigned bits = 0u;
        #pragma unroll 4
        for (int bit = 0; bit < 32; ++bit) {
            const int j = w * 32 + bit;
            if (j < NMS_TOP && j != i) {
                const float4 c = sBox[j];
                const float xx1 = fmaxf(a.x, c.x);
                const float yy1 = fmaxf(a.y, c.y);
                const float xx2 = fminf(a.z, c.z);
                const float yy2 = fminf(a.w, c.w);
                const float inter = fmaxf(xx2 - xx1, 0.f) * fmaxf(yy2 - yy1, 0.f);
                const float iou   = inter / (aa + sArea[j] - inter);
                if (!(iou < NMS_THR)) bits |= (1u << bit);
            }
        }
        sMat[i * K2_WORDS + w] = bits;
    }
    __syncthreads();

    // Serial greedy scan on a single wave32; no block barriers.
    if (tid < 32) {
        unsigned rem = 0u;   // removed bits for j in [tid*32, tid*32+32)
        for (int i = 0; i < NMS_TOP; ++i) {
            const unsigned remWord = (unsigned)__shfl((int)rem, i >> 5, 32);
            if (((remWord >> (i & 31)) & 1u) == 0u)      // box i survives
                rem |= sMat[i * K2_WORDS + tid];         // conflict-free row OR
        }
        sRem[tid] = rem;
    }
    __syncthreads();

    if (tid < NMS_TOP) {
        const bool r = ((sRem[tid >> 5] >> (tid & 31)) & 1u) != 0u;
        const size_t src = (size_t)b * NMS_TOP + tid;
        const float4 bx = sBox[tid];
        float4 o;
        o.x = r ? 0.f : bx.x;
        o.y = r ? 0.f : bx.y;
        o.z = r ? 0.f : bx.z;
        o.w = r ? 0.f : bx.w;
        *reinterpret_cast<float4*>(outReg + src * 4) = o;
        outCls[src] = r ? 0.f : wsCls[src];
    }
}

// ---------------------------------------------------------------------------
extern "C" void kernel_launch(void* const* d_in, const int* in_sizes, int n_in,
                              void* d_out, int out_size, void* d_ws, size_t ws_size,
                              hipStream_t stream) {
    (void)in_sizes; (void)n_in; (void)out_size; (void)ws_size;

    const float* reg = (const float*)d_in[0];   // (16, 20000, 4) f32
    const float* cls = (const float*)d_in[1];   // (16, 20000)    f32

    float* outReg = (float*)d_out;                        // (16,1000,4)
    float* outCls = outReg + (size_t)NMS_B * NMS_TOP * 4; // (16,1000)

    float* wsBox = (float*)d_ws;                          // (16,1000,4)
    float* wsCls = wsBox + (size_t)NMS_B * NMS_TOP * 4;   // (16,1000)

    dim3 g1((NMS_P + K1_SPAN - 1) / K1_SPAN, NMS_B);      // (10, 16)
    nms_topk_rank<<<g1, K1_THREADS, 0, stream>>>(reg, cls, wsBox, wsCls);

    nms_greedy<<<NMS_B, K2_THREADS, 0, stream>>>(wsBox, wsCls, outReg, outCls);
}